// PostprocessLayer_59596966199753
// MI455X (gfx1250) — compile-verified
//
#include <hip/hip_runtime.h>
#include <math.h>

#define N_CLASSES   80
#define MAX_BOXES_  20
#define OBJ_THR_    0.3f
#define NMS_THR_    0.45f
#define NMS_BLOCK   256
#define TILE        1792                       // boxes per LDS tile (7 * 256)
#define TILE_F      (TILE * 4)                 // 7168 floats per LDS buffer (28 KB)
#define LOADS_PER_THREAD (TILE / NMS_BLOCK)    // 7 x b128 per thread per staged chunk

typedef __attribute__((ext_vector_type(4))) unsigned uint4v;
typedef __attribute__((ext_vector_type(8))) unsigned uint8v;

__device__ __forceinline__ float sigmoidf_(float x) {
  return 1.0f / (1.0f + expf(-x));
}

// ---------------------------------------------------------------------------
// Phase 1: decode one YOLO level -> boxes[N][4] and class-major masked scores
// ---------------------------------------------------------------------------
__global__ void decode_level_kernel(const float* __restrict__ feat,
                                    const float* __restrict__ image_shape,
                                    float* __restrict__ boxes,     // [N*4]
                                    float* __restrict__ scores_t,  // [80*N]
                                    int H, int W, int count, int levelBase, int N,
                                    float aw0, float ah0, float aw1, float ah1,
                                    float aw2, float ah2,
                                    float input_h, float input_w)
{
  int n = blockIdx.x * blockDim.x + threadIdx.x;
  if (n >= count) return;
  int a    = n % 3;
  int cell = n / 3;
  int gx   = cell % W;
  int gy   = (cell / W) % H;
  const float* p = feat + (size_t)cell * (3 * (N_CLASSES + 5))
                        + (size_t)a * (N_CLASSES + 5);

  // letterbox correction constants (uniform)
  float imH = image_shape[0], imW = image_shape[1];
  float sm   = fminf(input_h / imH, input_w / imW);
  float newH = rintf(imH * sm), newW = rintf(imW * sm);   // round-half-even == jnp.round
  float offY = (input_h - newH) * 0.5f / input_h;
  float offX = (input_w - newW) * 0.5f / input_w;
  float scY  = input_h / newH;
  float scX  = input_w / newW;

  float bx = (sigmoidf_(p[0]) + (float)gx) / (float)W;
  float by = (sigmoidf_(p[1]) + (float)gy) / (float)H;
  float aw = (a == 0) ? aw0 : ((a == 1) ? aw1 : aw2);
  float ah = (a == 0) ? ah0 : ((a == 1) ? ah1 : ah2);
  float bw = expf(p[2]) * aw / input_w;
  float bh = expf(p[3]) * ah / input_h;

  float cy = (by - offY) * scY;
  float cx = (bx - offX) * scX;
  float hh = bh * scY;
  float ww = bw * scX;
  float y1 = (cy - 0.5f * hh) * imH;
  float x1 = (cx - 0.5f * ww) * imW;
  float y2 = (cy + 0.5f * hh) * imH;
  float x2 = (cx + 0.5f * ww) * imW;

  int g = levelBase + n;
  float4 bb; bb.x = y1; bb.y = x1; bb.z = y2; bb.w = x2;
  reinterpret_cast<float4*>(boxes)[g] = bb;

  float conf = sigmoidf_(p[4]);
  for (int c = 0; c < N_CLASSES; ++c) {
    float v = conf * sigmoidf_(p[5 + c]);
    scores_t[(size_t)c * N + g] = (v >= OBJ_THR_) ? v : -1.0f;   // coalesced along N
  }
}

// ---------------------------------------------------------------------------
// CDNA5 async global->LDS staging of a 7168-float chunk (ASYNCcnt-tracked).
// Every thread issues exactly LOADS_PER_THREAD b128 loads (EXEC kept full by
// clamping), so each wave's ASYNCcnt rises by exactly 7 per chunk.
// ---------------------------------------------------------------------------
__device__ __forceinline__ void async_stage_f32(const float* __restrict__ src,
                                                float* ldsBuf, int start4,
                                                int tid, int N)
{
  // low 32 bits of the generic pointer == LDS byte offset (LDS aperture rule)
  unsigned ldsBase = (unsigned)(unsigned long long)(void*)ldsBuf;
  const int gmax = (N - 4) & ~3;               // N is a multiple of 4 here
#pragma unroll
  for (int k = 0; k < LOADS_PER_THREAD; ++k) {
    int j  = tid + k * NMS_BLOCK;
    int g4 = start4 + j * 4;
    if (g4 > gmax) g4 = gmax;                  // stay inside the score row
    asm volatile("global_load_async_to_lds_b128 %0, %1, off"
                 :: "v"(ldsBase + (unsigned)(j * 16)), "v"(src + g4) : "memory");
  }
}

// ---------------------------------------------------------------------------
// CDNA5 Tensor Data Mover: DMA one 1-D box tile (TILE boxes = 7168 f32) into
// LDS with a single tensor_load_to_lds. D# per CDNA5 ISA:
//   group0: count=1 | lds_addr | global_addr(tile start) | type=2
//   group1: data_size=4B, tensor_dim0 = elements remaining (OOB reads -> 0),
//           tile_dim0 = 7168, everything else unused/zero.
// Issued by one wave; tracked with TENSORcnt.
// ---------------------------------------------------------------------------
__device__ __forceinline__ void tdm_load_boxes(const float* __restrict__ boxes,
                                               float* ldsBuf, int boxStart, int N)
{
  unsigned long long ga = (unsigned long long)(boxes + (size_t)boxStart * 4);
  unsigned lds    = (unsigned)(unsigned long long)(void*)ldsBuf;
  unsigned remain = (unsigned)((N - boxStart) * 4);      // f32 elements left
  uint4v g0 = { 1u,                                      // count = 1 (valid D#)
                lds,                                     // lds_addr
                (unsigned)ga,                            // global_addr[31:0]
                (unsigned)(ga >> 32) | (2u << 30) };     // global_addr[56:32]|type=2
  uint8v g1 = { 2u << 16,                                // data_size = 4 bytes
                (remain & 0xFFFFu) << 16,                // tensor_dim0[15:0]
                ((remain >> 16) & 0xFFFFu) | (1u << 16), // tensor_dim0[31:16]|tensor_dim1=1
                (unsigned)TILE_F << 16,                  // tile_dim0 = 7168
                0u,                                      // tile_dim1/2 unused
                (unsigned)TILE_F,                        // tensor_dim0_stride (unused, 1D)
                0u, 0u };
  asm volatile("tensor_load_to_lds %0, %1" :: "s"(g0), "s"(g1) : "memory");
}

// ---------------------------------------------------------------------------
// Phase 2: per-class sequential NMS (one workgroup per class, 8 x wave32)
// ---------------------------------------------------------------------------
__global__ __launch_bounds__(NMS_BLOCK)
void nms_kernel(const float* __restrict__ boxes,
                float* __restrict__ scores_t,
                float* __restrict__ out, int N)
{
  const int c    = blockIdx.x;
  float* s       = scores_t + (size_t)c * N;
  const int tid  = threadIdx.x;
  const int lane = tid & 31;
  const int wave = tid >> 5;

  // 2 x 28 KB buffers, time-shared: score chunks (argmax) / box tiles (IoU)
  __shared__ __align__(16) float tile[2][TILE_F];
  __shared__ float wv[NMS_BLOCK / 32];
  __shared__ int   wi[NMS_BLOCK / 32];
  __shared__ float sBest[6];                   // y1,x1,y2,x2, area, value

  float* outBoxes   = out;                                      // 80*20*4
  float* outClasses = out + N_CLASSES * MAX_BOXES_ * 4;         // 80*20
  float* outScores  = outClasses + N_CLASSES * MAX_BOXES_;      // 80*20

  const int numTilesB = (N + TILE - 1) / TILE;        // box tiles
  const int numTilesS = (N + TILE_F - 1) / TILE_F;    // score chunks

  for (int it = 0; it < MAX_BOXES_; ++it) {
    // ---- argmax over s[0..N): async-LDS staged, first-occurrence tie-break ----
    float bv = -3.402823466e38f; int bi = 0;
    async_stage_f32(s, &tile[0][0], 0, tid, N);       // prime the pipeline
    for (int t = 0; t < numTilesS; ++t) {
      const int cur = t & 1;
      if (t + 1 < numTilesS) {
        async_stage_f32(s, &tile[cur ^ 1][0], (t + 1) * TILE_F, tid, N);
        asm volatile("s_wait_asynccnt 0x7" ::: "memory");  // chunk t done, t+1 in flight
      } else {
        asm volatile("s_wait_asynccnt 0x0" ::: "memory");
      }
      __syncthreads();
      const int base = t * TILE_F;
#pragma unroll
      for (int k = 0; k < LOADS_PER_THREAD; ++k) {
        const int j4 = (tid + k * NMS_BLOCK) * 4;
        const int i  = base + j4;
        const float4 v4 = *reinterpret_cast<const float4*>(&tile[cur][j4]);
        // in-order within the quad => strict '>' keeps earliest index
        if (i + 0 < N && v4.x > bv) { bv = v4.x; bi = i + 0; }
        if (i + 1 < N && v4.y > bv) { bv = v4.y; bi = i + 1; }
        if (i + 2 < N && v4.z > bv) { bv = v4.z; bi = i + 2; }
        if (i + 3 < N && v4.w > bv) { bv = v4.w; bi = i + 3; }
      }
      __syncthreads();                         // done reading buf[cur] before reuse
    }
#pragma unroll
    for (int off = 16; off > 0; off >>= 1) {   // wave32 reduction
      float ov = __shfl_xor(bv, off, 32);
      int   oi = __shfl_xor(bi, off, 32);
      if (ov > bv || (ov == bv && oi < bi)) { bv = ov; bi = oi; }
    }
    if (lane == 0) { wv[wave] = bv; wi[wave] = bi; }
    __syncthreads();
    if (tid == 0) {
      float fbv = wv[0]; int fbi = wi[0];
      for (int w = 1; w < NMS_BLOCK / 32; ++w)
        if (wv[w] > fbv || (wv[w] == fbv && wi[w] < fbi)) { fbv = wv[w]; fbi = wi[w]; }
      const float4 bb = reinterpret_cast<const float4*>(boxes)[fbi];
      const bool valid = fbv > 0.0f;
      const int  o = c * MAX_BOXES_ + it;
      outBoxes[o * 4 + 0] = valid ? bb.x : 0.0f;
      outBoxes[o * 4 + 1] = valid ? bb.y : 0.0f;
      outBoxes[o * 4 + 2] = valid ? bb.z : 0.0f;
      outBoxes[o * 4 + 3] = valid ? bb.w : 0.0f;
      outClasses[o] = (float)c;
      outScores[o]  = valid ? fbv : -1.0f;
      sBest[0] = bb.x; sBest[1] = bb.y; sBest[2] = bb.z; sBest[3] = bb.w;
      sBest[4] = (bb.z - bb.x) * (bb.w - bb.y);
      sBest[5] = fbv;
      if (valid) s[fbi] = -1.0f;               // .at[i].set(-1)
      __threadfence_block();
    }
    __syncthreads();

    if (sBest[5] > 0.0f) {                     // uniform branch (shared value)
      const float qy1 = sBest[0], qx1 = sBest[1], qy2 = sBest[2], qx2 = sBest[3];
      const float qa  = sBest[4];

      if (wave == 0) tdm_load_boxes(boxes, &tile[0][0], 0, N);   // prime TDM
      for (int t = 0; t < numTilesB; ++t) {
        const int cur = t & 1;
        if (wave == 0) {
          if (t + 1 < numTilesB) {
            tdm_load_boxes(boxes, &tile[cur ^ 1][0], (t + 1) * TILE, N);
            __builtin_amdgcn_s_wait_tensorcnt(1);   // tile t done, t+1 in flight
          } else {
            __builtin_amdgcn_s_wait_tensorcnt(0);
          }
        }
        __syncthreads();                       // whole tile visible to all waves
        const int base = t * TILE;
#pragma unroll
        for (int k = 0; k < LOADS_PER_THREAD; ++k) {
          const int j = tid + k * NMS_BLOCK;
          const int i = base + j;
          if (i < N) {
            float y1 = tile[cur][j * 4 + 0];
            float x1 = tile[cur][j * 4 + 1];
            float y2 = tile[cur][j * 4 + 2];
            float x2 = tile[cur][j * 4 + 3];
            float iy1 = fmaxf(qy1, y1), ix1 = fmaxf(qx1, x1);
            float iy2 = fminf(qy2, y2), ix2 = fminf(qx2, x2);
            float ih = fmaxf(iy2 - iy1, 0.0f), iw = fmaxf(ix2 - ix1, 0.0f);
            float inter = ih * iw;
            float area  = (y2 - y1) * (x2 - x1);
            float iou   = inter / (qa + area - inter + 1e-9f);
            if (iou > NMS_THR_) s[i] = -1.0f;
          }
        }
        __syncthreads();                       // done reading buf[cur] before reuse
      }
      __threadfence_block();
      __syncthreads();
    }
  }
}

// ---------------------------------------------------------------------------
extern "C" void kernel_launch(void* const* d_in, const int* in_sizes, int n_in,
                              void* d_out, int out_size, void* d_ws, size_t ws_size,
                              hipStream_t stream) {
  (void)n_in; (void)out_size; (void)ws_size;
  const float* feat0 = (const float*)d_in[0];   // (B,19,19,255)
  const float* feat1 = (const float*)d_in[1];   // (B,38,38,255)
  const float* feat2 = (const float*)d_in[2];   // (B,76,76,255)
  const float* ishp  = (const float*)d_in[3];   // (2,)
  float* out = (float*)d_out;

  const int B  = in_sizes[0] / (19 * 19 * 255);
  const int c0 = B * 19 * 19 * 3;
  const int c1 = B * 38 * 38 * 3;
  const int c2 = B * 76 * 76 * 3;
  const int N  = c0 + c1 + c2;                  // 90972 for B=4
  const float inH = 19.0f * 32.0f, inW = 19.0f * 32.0f;

  float* boxes  = (float*)d_ws;                 // N*4 floats
  float* scores = boxes + (size_t)N * 4;        // 80*N floats (class-major)

  const int TB = 256;
  // level0: anchors mask [6,7,8]
  decode_level_kernel<<<(c0 + TB - 1) / TB, TB, 0, stream>>>(
      feat0, ishp, boxes, scores, 19, 19, c0, 0, N,
      116.f, 90.f, 156.f, 198.f, 373.f, 326.f, inH, inW);
  // level1: anchors mask [3,4,5]
  decode_level_kernel<<<(c1 + TB - 1) / TB, TB, 0, stream>>>(
      feat1, ishp, boxes, scores, 38, 38, c1, c0, N,
      30.f, 61.f, 62.f, 45.f, 59.f, 119.f, inH, inW);
  // level2: anchors mask [0,1,2]
  decode_level_kernel<<<(c2 + TB - 1) / TB, TB, 0, stream>>>(
      feat2, ishp, boxes, scores, 76, 76, c2, c0 + c1, N,
      10.f, 13.f, 16.f, 30.f, 33.f, 23.f, inH, inW);

  // one workgroup per class
  nms_kernel<<<N_CLASSES, NMS_BLOCK, 0, stream>>>(boxes, scores, out, N);
}